// FluidAutoencoder_10608569221424
// MI455X (gfx1250) — compile-verified
//
#include <hip/hip_runtime.h>
#include <hip/hip_bf16.h>
#include <math.h>

// ---------------------------------------------------------------------------
// FluidAutoencoder pipeline for MI455X (gfx1250, wave32).
//   conv1 -> conv2 -> GRU xproj (WMMA f32 16x16x4) -> GRU scan -> deconv1 -> deconv2
// Convs are bandwidth-bound (2/8/16 channels) -> direct VALU NHWC kernels.
// GRU xproj is the only GEMM-shaped op -> V_WMMA_F32_16X16X4_F32 (fp32-exact).
// ---------------------------------------------------------------------------

typedef __attribute__((ext_vector_type(2))) float v2f;
typedef __attribute__((ext_vector_type(8))) float v8f;

#define BN_RS 0.99950037462234706f /* rsqrt(1.001) computed at f32 precision below */

__device__ __forceinline__ float leaky(float x) { return x >= 0.f ? x : 0.2f * x; }
__device__ __forceinline__ float bnscale(float g) { return g * rsqrtf(1.001f); }

// ------------------------- conv1: (256,160,160,2)->(256,40,40,8), s4, SAME(pad 1,2)
__global__ void k_conv1(const float* __restrict__ in, const float* __restrict__ w,
                        const float* __restrict__ bias, const float* __restrict__ gamma,
                        const float* __restrict__ beta, float* __restrict__ out) {
  int idx = blockIdx.x * blockDim.x + threadIdx.x;
  const int N = 256 * 40 * 40 * 8;
  if (idx >= N) return;
  int co = idx & 7;
  int t = idx >> 3;
  int ow = t % 40; t /= 40;
  int oh = t % 40; int b = t / 40;
  int ih0 = oh * 4 - 1, iw0 = ow * 4 - 1;
  float acc = bias[co];
  for (int kh = 0; kh < 7; ++kh) {
    int ih = ih0 + kh; if ((unsigned)ih >= 160u) continue;
    for (int kw = 0; kw < 7; ++kw) {
      int iw = iw0 + kw; if ((unsigned)iw >= 160u) continue;
      const float* ip = in + ((b * 160 + ih) * 160 + iw) * 2;
      const float* wp = w + ((kh * 7 + kw) * 2) * 8 + co;  // HWIO
      acc += ip[0] * wp[0] + ip[1] * wp[8];
    }
  }
  out[idx] = leaky(acc) * bnscale(gamma[co]) + beta[co];
}

// ------------------------- conv2: (256,40,40,8)->(256,10,10,16), s4, SAME(pad 1,2)
__global__ void k_conv2(const float* __restrict__ in, const float* __restrict__ w,
                        const float* __restrict__ bias, const float* __restrict__ gamma,
                        const float* __restrict__ beta, float* __restrict__ out) {
  int idx = blockIdx.x * blockDim.x + threadIdx.x;
  const int N = 256 * 10 * 10 * 16;
  if (idx >= N) return;
  int co = idx & 15;
  int t = idx >> 4;
  int ow = t % 10; t /= 10;
  int oh = t % 10; int b = t / 10;
  int ih0 = oh * 4 - 1, iw0 = ow * 4 - 1;
  float acc = bias[co];
  for (int kh = 0; kh < 7; ++kh) {
    int ih = ih0 + kh; if ((unsigned)ih >= 40u) continue;
    for (int kw = 0; kw < 7; ++kw) {
      int iw = iw0 + kw; if ((unsigned)iw >= 40u) continue;
      const float* ip = in + ((b * 40 + ih) * 40 + iw) * 8;
      const float* wp = w + ((kh * 7 + kw) * 8) * 16 + co;  // HWIO
      #pragma unroll
      for (int ci = 0; ci < 8; ++ci) acc += ip[ci] * wp[ci * 16];
    }
  }
  out[idx] = leaky(acc) * bnscale(gamma[co]) + beta[co];
}

// ------------------------- GRU xproj: per channel c, [256x100] @ [100x300] + bx
// frames[c][t][d] = x2[t*1600 + d*16 + c]; wx[c][d][k]; out xproj[c][t][k] (stride 300)
// One wave per 16x16 D-tile. M=256 (16 tiles), N=300 (19 tiles, masked), K=100 (25x4).
__global__ void k_gru_xproj_wmma(const float* __restrict__ x2,
                                 const float* __restrict__ wx,
                                 const float* __restrict__ gb,
                                 float* __restrict__ xproj) {
  const int NT = 19;
  int wid  = blockIdx.x * (blockDim.x >> 5) + (threadIdx.x >> 5);
  int c    = wid / (16 * NT);
  int rem  = wid % (16 * NT);
  int mt   = rem / NT, nt = rem % NT;
  int lane = threadIdx.x & 31;
  int half = lane >> 4;        // 0: lanes 0-15, 1: lanes 16-31
  int l15  = lane & 15;
  int Mb = mt * 16, Nb = nt * 16;
  int m  = Mb + l15;
  int n  = Nb + l15;
  int ncl = n < 300 ? n : 299;            // clamp, mask with 0.f (keep EXEC all-1s)
  float nmask = (n < 300) ? 1.f : 0.f;

  const float* pA = x2 + m * 1600 + c;            // A[m][k] at pA[k*16]
  const float* pB = wx + c * 30000 + ncl;         // B[k][n] at pB[k*300]

  v8f acc = {};
  for (int k0 = 0; k0 < 100; k0 += 4) {
    int ka = k0 + 2 * half;                       // per-half-wave K pair (ISA A/B layout)
    v2f a, b;
    a.x = pA[ka * 16];
    a.y = pA[(ka + 1) * 16];
    b.x = pB[ka * 300] * nmask;
    b.y = pB[(ka + 1) * 300] * nmask;
    // D = A*B + C, fp32 WMMA: v_wmma_f32_16x16x4_f32
    acc = __builtin_amdgcn_wmma_f32_16x16x4_f32(false, a, false, b,
                                                (short)0, acc, false, false);
  }
  if (n < 300) {
    float bxv = gb[c * 600 + n];                  // bx[c][n]
    float* po = xproj + (c * 256 + Mb + 8 * half) * 300 + n;  // row M=r+8*half
    #pragma unroll
    for (int r = 0; r < 8; ++r) po[r * 300] = acc[r] + bxv;
  }
}

// ------------------------- GRU scan: 16 blocks (one per channel), 320 threads.
// rec[k] = h . wh[c][:,k] + bh; gates; writes rnn NHWC (256,10,10,16) directly.
__global__ void k_gru_scan(const float* __restrict__ xproj,
                           const float* __restrict__ wh,
                           const float* __restrict__ gb,
                           float* __restrict__ rnn) {
  __shared__ float h_s[100];
  __shared__ float rec_s[300];
  int c = blockIdx.x;
  int j = threadIdx.x;
  const float* whc = wh + c * 30000;
  const float* bhc = gb + c * 600 + 300;
  const float* xc  = xproj + c * 256 * 300;
  if (j < 100) h_s[j] = 0.f;
  __syncthreads();
  for (int t = 0; t < 256; ++t) {
    if (j < 300) {
      float s = bhc[j];
      for (int d = 0; d < 100; ++d) s += h_s[d] * whc[d * 300 + j];
      rec_s[j] = s;
    }
    __syncthreads();
    float hn = 0.f;
    if (j < 100) {
      const float* xt = xc + t * 300;
      float z  = 1.f / (1.f + expf(-(xt[j]       + rec_s[j])));
      float r  = 1.f / (1.f + expf(-(xt[j + 100] + rec_s[j + 100])));
      float hh = tanhf(xt[j + 200] + r * rec_s[j + 200]);
      hn = z * h_s[j] + (1.f - z) * hh;
      rnn[t * 1600 + j * 16 + c] = hn;   // rnn[b=t][p=j][c]
    }
    __syncthreads();
    if (j < 100) h_s[j] = hn;
    __syncthreads();
  }
}

// ------------------------- deconv1: (256,10,10,16)->(256,40,40,8), s4 transpose,
// SAME => dilate by 4, pad (5,4), correlate 7x7 (no kernel flip), leaky + bn1.
__global__ void k_deconv1(const float* __restrict__ in, const float* __restrict__ w,
                          const float* __restrict__ bias, const float* __restrict__ gamma,
                          const float* __restrict__ beta, float* __restrict__ out) {
  int idx = blockIdx.x * blockDim.x + threadIdx.x;
  const int N = 256 * 40 * 40 * 8;
  if (idx >= N) return;
  int co = idx & 7;
  int t = idx >> 3;
  int ow = t % 40; t /= 40;
  int oh = t % 40; int b = t / 40;
  float acc = bias[co];
  for (int th = 0; th < 7; ++th) {
    int nh = oh + th - 5;
    if (nh < 0 || (nh & 3)) continue;
    int ih = nh >> 2; if (ih >= 10) continue;
    for (int tw = 0; tw < 7; ++tw) {
      int nw = ow + tw - 5;
      if (nw < 0 || (nw & 3)) continue;
      int iw = nw >> 2; if (iw >= 10) continue;
      const float* ip = in + ((b * 10 + ih) * 10 + iw) * 16;
      const float* wp = w + ((th * 7 + tw) * 16) * 8 + co;   // HWIO, I=16
      #pragma unroll
      for (int ci = 0; ci < 16; ++ci) acc += ip[ci] * wp[ci * 8];
    }
  }
  out[idx] = leaky(acc) * bnscale(gamma[co]) + beta[co];
}

// ------------------------- deconv2: (256,40,40,8)->(256,160,160,2), leaky, *gs
__global__ void k_deconv2(const float* __restrict__ in, const float* __restrict__ w,
                          const float* __restrict__ bias, const float* __restrict__ gs,
                          float* __restrict__ out) {
  int idx = blockIdx.x * blockDim.x + threadIdx.x;
  const int N = 256 * 160 * 160 * 2;
  if (idx >= N) return;
  int co = idx & 1;
  int t = idx >> 1;
  int ow = t % 160; t /= 160;
  int oh = t % 160; int b = t / 160;
  float acc = bias[co];
  for (int th = 0; th < 7; ++th) {
    int nh = oh + th - 5;
    if (nh < 0 || (nh & 3)) continue;
    int ih = nh >> 2; if (ih >= 40) continue;
    for (int tw = 0; tw < 7; ++tw) {
      int nw = ow + tw - 5;
      if (nw < 0 || (nw & 3)) continue;
      int iw = nw >> 2; if (iw >= 40) continue;
      const float* ip = in + ((b * 40 + ih) * 40 + iw) * 8;
      const float* wp = w + ((th * 7 + tw) * 8) * 2 + co;    // HWIO, I=8
      #pragma unroll
      for (int ci = 0; ci < 8; ++ci) acc += ip[ci] * wp[ci * 2];
    }
  }
  out[idx] = gs[0] * leaky(acc);
}

extern "C" void kernel_launch(void* const* d_in, const int* in_sizes, int n_in,
                              void* d_out, int out_size, void* d_ws, size_t ws_size,
                              hipStream_t stream) {
  (void)in_sizes; (void)n_in; (void)out_size; (void)ws_size;
  const float* lo  = (const float*)d_in[0];
  const float* c1k = (const float*)d_in[1];
  const float* c1b = (const float*)d_in[2];
  const float* g1  = (const float*)d_in[3];
  const float* b1  = (const float*)d_in[4];
  const float* c2k = (const float*)d_in[5];
  const float* c2b = (const float*)d_in[6];
  const float* g2  = (const float*)d_in[7];
  const float* b2  = (const float*)d_in[8];
  const float* wx  = (const float*)d_in[9];
  const float* wh  = (const float*)d_in[10];
  const float* gb  = (const float*)d_in[11];
  const float* d1k = (const float*)d_in[12];
  const float* d1b = (const float*)d_in[13];
  const float* d2k = (const float*)d_in[14];
  const float* d2b = (const float*)d_in[15];
  const float* gs  = (const float*)d_in[16];

  float* ws  = (float*)d_ws;
  float* x1  = ws;               // (256,40,40,8)   = 3,276,800
  float* x2  = x1 + 3276800;     // (256,10,10,16)  =   409,600
  float* xp  = x2 + 409600;      // (16,256,300)    = 1,228,800
  float* rnn = xp + 1228800;     // (256,10,10,16)  =   409,600
  float* y1  = rnn + 409600;     // (256,40,40,8)   = 3,276,800
  float* out = (float*)d_out;    // (256,160,160,2)

  { int n = 256 * 40 * 40 * 8;
    k_conv1<<<(n + 255) / 256, 256, 0, stream>>>(lo, c1k, c1b, g1, b1, x1); }
  { int n = 256 * 10 * 10 * 16;
    k_conv2<<<(n + 255) / 256, 256, 0, stream>>>(x1, c2k, c2b, g2, b2, x2); }
  // 16 chans * 16 Mtiles * 19 Ntiles = 4864 waves; 4 waves/block -> 1216 blocks
  k_gru_xproj_wmma<<<1216, 128, 0, stream>>>(x2, wx, gb, xp);
  k_gru_scan<<<16, 320, 0, stream>>>(xp, wh, gb, rnn);
  { int n = 256 * 40 * 40 * 8;
    k_deconv1<<<(n + 255) / 256, 256, 0, stream>>>(rnn, d1k, d1b, g1, b1, y1); }
  { int n = 256 * 160 * 160 * 2;
    k_deconv2<<<(n + 255) / 256, 256, 0, stream>>>(y1, d2k, d2b, gs, out); }
}